// SPDLogisticClassifier_78606491451806
// MI455X (gfx1250) — compile-verified
//
#include <hip/hip_runtime.h>
#include <hip/hip_bf16.h>
#include <math.h>

// ---------------------------------------------------------------------------
// SPD logistic classifier, MI455X (gfx1250, wave32).
// All heavy math (matrix exp/log/sqrt of 16x16 SPD) expressed as 16x16 f32
// matmuls on V_WMMA_F32_16X16X4_F32. Global->LDS staging uses CDNA5 async
// loads (GLOBAL_LOAD_ASYNC_TO_LDS_B128 + s_wait_asynccnt).
// One wave handles one (n, c) pair.
// ---------------------------------------------------------------------------

#define DD      16
#define CC      16
#define NN      20000
#define UT      136
#define WPB     4          // waves per block in main kernel

typedef __attribute__((ext_vector_type(2))) float v2f;
typedef __attribute__((ext_vector_type(8))) float v8f;

#define WBAR() __builtin_amdgcn_wave_barrier()

// --- CDNA5 async global->LDS copy of one 1KB tile (256 floats) -------------
// Each lane moves 16B at [gsrc + lane*16] and [gsrc + lane*16 + 512]; the
// immediate offset applies to both LDS and global addresses (ISA 08 §4.4).
// Tracked by ASYNCcnt; caller waits with s_wait_asynccnt.
__device__ __forceinline__ void async_copy_1kb(const float* __restrict__ gsrc,
                                               float* __restrict__ lds_dst,
                                               unsigned lane) {
  const void* gp = (const void*)(gsrc + lane * 4u);              // 16B/lane
  unsigned ldsoff = (unsigned)(uintptr_t)lds_dst + lane * 16u;   // LDS byte addr
  asm volatile(
      "global_load_async_to_lds_b128 %0, %1, off\n\t"
      "global_load_async_to_lds_b128 %0, %1, off offset:512"
      :
      : "v"(ldsoff), "v"(gp)
      : "memory");
}

__device__ __forceinline__ void async_wait0() {
  asm volatile("s_wait_asynccnt 0x0" ::: "memory");
}

// --- wave-level 16x16x16 f32 matmul: D = X * Y, X/Y row-major in LDS -------
// WMMA f32 16x16x4 layouts (ISA 7.12.2):
//   A (16x4): lane L: M = L%16, holds K = 2*(L/16)+{0,1}
//   B (4x16): lane L: N = L%16, holds K = 2*(L/16)+{0,1}
//   C/D     : lane L: N = L%16, VGPR r holds M = r + 8*(L/16)
__device__ __forceinline__ v8f wave_mm16(const float* __restrict__ X,
                                         const float* __restrict__ Y,
                                         unsigned lane) {
  const unsigned half = lane >> 4;
  const unsigned mn   = lane & 15u;
  v8f acc = {0.f, 0.f, 0.f, 0.f, 0.f, 0.f, 0.f, 0.f};
#pragma unroll
  for (int kb = 0; kb < 4; ++kb) {
    const int k0 = kb * 4 + (int)(half << 1);
    v2f a, b;
    a[0] = X[mn * 16 + k0];
    a[1] = X[mn * 16 + k0 + 1];
    b[0] = Y[k0 * 16 + mn];
    b[1] = Y[(k0 + 1) * 16 + mn];
    acc = __builtin_amdgcn_wmma_f32_16x16x4_f32(false, a, false, b,
                                                (short)0, acc, false, false);
  }
  return acc;
}

// store C/D fragment to row-major 16x16 LDS buffer
__device__ __forceinline__ void frag_store(float* __restrict__ M, v8f f,
                                           unsigned lane) {
  const unsigned half = lane >> 4;
  const unsigned n    = lane & 15u;
#pragma unroll
  for (int r = 0; r < 8; ++r) M[(r + 8 * half) * 16 + n] = f[r];
}

__device__ __forceinline__ float wave_frob(const float* __restrict__ M,
                                           unsigned lane) {
  float s = 0.f;
#pragma unroll
  for (int i = 0; i < 8; ++i) {
    float t = M[lane * 8 + i];
    s += t * t;
  }
#pragma unroll
  for (int m = 16; m >= 1; m >>= 1) s += __shfl_xor(s, m, 32);
  return sqrtf(s);
}

// B <- sqrtm(B) via Frobenius-scaled coupled Newton-Schulz.
__device__ __forceinline__ void wave_sqrtm(float* B, float* Y, float* Z,
                                           float* T, unsigned lane,
                                           int iters) {
  const float c  = wave_frob(B, lane);
  const float rc = 1.f / c;
#pragma unroll
  for (int i = 0; i < 8; ++i) {
    const int idx = (int)lane * 8 + i;
    Y[idx] = B[idx] * rc;
    Z[idx] = ((idx >> 4) == (idx & 15)) ? 1.f : 0.f;
  }
  WBAR();
#pragma unroll 1
  for (int it = 0; it < iters; ++it) {
    v8f zy = wave_mm16(Z, Y, lane);
    frag_store(T, zy, lane);
    WBAR();
#pragma unroll
    for (int i = 0; i < 8; ++i) {
      const int idx = (int)lane * 8 + i;
      const float dg = ((idx >> 4) == (idx & 15)) ? 1.5f : 0.f;
      T[idx] = dg - 0.5f * T[idx];
    }
    WBAR();
    v8f yn = wave_mm16(Y, T, lane);
    v8f zn = wave_mm16(T, Z, lane);
    frag_store(Y, yn, lane);
    frag_store(Z, zn, lane);
    WBAR();
  }
  const float sc = sqrtf(c);
#pragma unroll
  for (int i = 0; i < 8; ++i) {
    const int idx = (int)lane * 8 + i;
    B[idx] = Y[idx] * sc;
  }
  WBAR();
}

// ---------------------------------------------------------------------------
// Kernel 1: per-class prep. 1 wave per class: x = expm(sym(means_c)) via
// scaling+Taylor+squaring, then coupled NS sqrt -> sqrt_x, inv_sqrt_x.
// ---------------------------------------------------------------------------
__global__ __launch_bounds__(128) void spd_class_prep(
    const float* __restrict__ means, float* __restrict__ sqrt_x,
    float* __restrict__ inv_sqrt_x) {
  __shared__ __align__(16) float buf[4][6][256];
  const unsigned wave = threadIdx.x >> 5;
  const unsigned lane = threadIdx.x & 31u;
  const int c = (int)(blockIdx.x * 4 + wave);
  if (c >= CC) return;

  float* M  = buf[wave][0];
  float* T  = buf[wave][1];
  float* P  = buf[wave][2];
  float* Y  = buf[wave][3];
  float* Z  = buf[wave][4];
  float* TT = buf[wave][5];

  // M = sym(means[c]) * 2^-6 ; T = I
#pragma unroll
  for (int i = 0; i < 8; ++i) {
    const int idx = (int)lane * 8 + i;
    const int r = idx >> 4, q = idx & 15;
    M[idx] = 0.5f * (means[c * 256 + r * 16 + q] +
                     means[c * 256 + q * 16 + r]) * (1.f / 64.f);
    T[idx] = (r == q) ? 1.f : 0.f;
  }
  WBAR();

  // Horner Taylor, 12 terms: T = I + M/k * T  (fully unrolled so 1/k folds
  // to a literal -- avoids the v_div_scale/v_div_fmas sequence in the loop)
#pragma unroll
  for (int k = 12; k >= 1; --k) {
    v8f p = wave_mm16(M, T, lane);
    frag_store(P, p, lane);
    WBAR();
    const float rk = 1.f / (float)k;
#pragma unroll
    for (int i = 0; i < 8; ++i) {
      const int idx = (int)lane * 8 + i;
      T[idx] = (((idx >> 4) == (idx & 15)) ? 1.f : 0.f) + P[idx] * rk;
    }
    WBAR();
  }
  // 6 squarings
#pragma unroll 1
  for (int s = 0; s < 6; ++s) {
    v8f q = wave_mm16(T, T, lane);
    frag_store(P, q, lane);
    WBAR();
#pragma unroll
    for (int i = 0; i < 8; ++i) {
      const int idx = (int)lane * 8 + i;
      T[idx] = P[idx];
    }
    WBAR();
  }

  // coupled NS: Y -> sqrt(x/c), Z -> invsqrt(x/c)
  const float cF = wave_frob(T, lane);
  const float rc = 1.f / cF;
#pragma unroll
  for (int i = 0; i < 8; ++i) {
    const int idx = (int)lane * 8 + i;
    Y[idx] = T[idx] * rc;
    Z[idx] = ((idx >> 4) == (idx & 15)) ? 1.f : 0.f;
  }
  WBAR();
#pragma unroll 1
  for (int it = 0; it < 10; ++it) {
    v8f zy = wave_mm16(Z, Y, lane);
    frag_store(TT, zy, lane);
    WBAR();
#pragma unroll
    for (int i = 0; i < 8; ++i) {
      const int idx = (int)lane * 8 + i;
      const float dg = ((idx >> 4) == (idx & 15)) ? 1.5f : 0.f;
      TT[idx] = dg - 0.5f * TT[idx];
    }
    WBAR();
    v8f yn = wave_mm16(Y, TT, lane);
    v8f zn = wave_mm16(TT, Z, lane);
    frag_store(Y, yn, lane);
    frag_store(Z, zn, lane);
    WBAR();
  }
  const float sc  = sqrtf(cF);
  const float isc = 1.f / sc;
#pragma unroll
  for (int i = 0; i < 8; ++i) {
    const int idx = (int)lane * 8 + i;
    sqrt_x[c * 256 + idx]     = Y[idx] * sc;
    inv_sqrt_x[c * 256 + idx] = Z[idx] * isc;
  }
}

// ---------------------------------------------------------------------------
// Kernel 2: sig = expm(sym(sigma)) per class (136x136). One workgroup per
// class, matrices in global scratch. Negligible fraction of total work.
// ---------------------------------------------------------------------------
__global__ __launch_bounds__(256) void spd_sigma_expm(
    const float* __restrict__ sigma, float* __restrict__ sig_out,
    float* __restrict__ scratch) {
  const int c   = (int)blockIdx.x;
  const int tid = (int)threadIdx.x;
  float* A0 = scratch + (size_t)c * 3u * UT * UT;  // scaled sym input
  float* A1 = A0 + UT * UT;                        // accumulator T
  float* A2 = A1 + UT * UT;                        // product
  const float* Sin = sigma + (size_t)c * UT * UT;

  for (int idx = tid; idx < UT * UT; idx += 256) {
    const int i = idx / UT, j = idx % UT;
    A0[idx] = 0.5f * (Sin[i * UT + j] + Sin[j * UT + i]) * (1.f / 64.f);
    A1[idx] = (i == j) ? 1.f : 0.f;
  }
  __threadfence();
  __syncthreads();

  for (int k = 12; k >= 1; --k) {
    for (int idx = tid; idx < UT * UT; idx += 256) {
      const int i = idx / UT, j = idx % UT;
      float s = 0.f;
      for (int kk = 0; kk < UT; ++kk)
        s = fmaf(A0[i * UT + kk], A1[kk * UT + j], s);
      A2[idx] = s;
    }
    __threadfence();
    __syncthreads();
    const float rk = 1.f / (float)k;
    for (int idx = tid; idx < UT * UT; idx += 256) {
      const int i = idx / UT, j = idx % UT;
      A1[idx] = ((i == j) ? 1.f : 0.f) + A2[idx] * rk;
    }
    __threadfence();
    __syncthreads();
  }
  for (int s = 0; s < 6; ++s) {
    for (int idx = tid; idx < UT * UT; idx += 256) {
      const int i = idx / UT, j = idx % UT;
      float acc = 0.f;
      for (int kk = 0; kk < UT; ++kk)
        acc = fmaf(A1[i * UT + kk], A1[kk * UT + j], acc);
      A2[idx] = acc;
    }
    __threadfence();
    __syncthreads();
    for (int idx = tid; idx < UT * UT; idx += 256) A1[idx] = A2[idx];
    __threadfence();
    __syncthreads();
  }
  for (int idx = tid; idx < UT * UT; idx += 256)
    sig_out[(size_t)c * UT * UT + idx] = A1[idx];
}

// ---------------------------------------------------------------------------
// Kernel 3: main. One wave per (n,c) pair.
//   m      = isx_c * A_n * isx_c                       (WMMA)
//   logm   = 32 * logseries( (NS-sqrt)^5 (m) )         (WMMA)
//   u      = sx_c * logm * sx_c                        (WMMA)
//   out    = -0.5 * triu(u)^T Sig_c triu(u) + bias_c   (VALU + L2 streams)
// ---------------------------------------------------------------------------
__global__ __launch_bounds__(128) void spd_main(
    const float* __restrict__ node_feats, const float* __restrict__ sqrt_x,
    const float* __restrict__ inv_sqrt_x, const float* __restrict__ sig,
    const float* __restrict__ bias, float* __restrict__ out) {
  __shared__ __align__(16) float buf[WPB][6][256];
  __shared__ float vbuf[WPB][UT];
  __shared__ int rows_s[UT];
  __shared__ int cols_s[UT];

  if (threadIdx.x == 0) {
    int p = 0;
    for (int i = 0; i < DD; ++i)
      for (int j = i; j < DD; ++j) {
        rows_s[p] = i;
        cols_s[p] = j;
        ++p;
      }
  }
  __syncthreads();

  const unsigned wave = threadIdx.x >> 5;
  const unsigned lane = threadIdx.x & 31u;
  const unsigned pair = blockIdx.x * WPB + wave;  // N*C divisible by WPB
  const int n = (int)(pair / CC);
  const int c = (int)(pair % CC);

  float* B  = buf[wave][0];
  float* Y  = buf[wave][1];
  float* Z  = buf[wave][2];
  float* T  = buf[wave][3];
  float* W1 = buf[wave][4];
  float* W2 = buf[wave][5];

  const float* Sg = sig + (size_t)c * UT * UT;
  // touch the Sig_c block early (spread lanes across its 74KB; L2 resident)
  __builtin_prefetch(Sg + (size_t)lane * 578, 0, 1);

  // async global->LDS: inv_sqrt_x[c] -> W1, node_feats[n] -> W2
  async_copy_1kb(inv_sqrt_x + (size_t)c * 256, W1, lane);
  async_copy_1kb(node_feats + (size_t)n * 256, W2, lane);
  async_wait0();
  WBAR();

  // m = W1 * W2 * W1
  v8f t0 = wave_mm16(W1, W2, lane);
  frag_store(T, t0, lane);
  WBAR();
  v8f m0 = wave_mm16(T, W1, lane);
  frag_store(B, m0, lane);
  WBAR();

  // B <- m^(1/32) via 5 NS square roots
#pragma unroll 1
  for (int lvl = 0; lvl < 5; ++lvl) wave_sqrtm(B, Y, Z, T, lane, 6);

  // E = B - I; log(B) ~= E - E^2/2 + E^3/3 - E^4/4 ; scale by 32
#pragma unroll
  for (int i = 0; i < 8; ++i) {
    const int idx = (int)lane * 8 + i;
    B[idx] -= ((idx >> 4) == (idx & 15)) ? 1.f : 0.f;
  }
  WBAR();
  v8f e2 = wave_mm16(B, B, lane);
  frag_store(T, e2, lane);
  WBAR();
  v8f e3 = wave_mm16(T, B, lane);
  frag_store(W2, e3, lane);
  v8f e4 = wave_mm16(T, T, lane);
  frag_store(Y, e4, lane);
  WBAR();
#pragma unroll
  for (int i = 0; i < 8; ++i) {
    const int idx = (int)lane * 8 + i;
    B[idx] = 32.f * (B[idx] - 0.5f * T[idx] + (1.f / 3.f) * W2[idx] -
                     0.25f * Y[idx]);
  }
  WBAR();

  // u = sqrt_x * logm * sqrt_x  (async-load sqrt_x into W1, now free)
  async_copy_1kb(sqrt_x + (size_t)c * 256, W1, lane);
  async_wait0();
  WBAR();
  v8f t1 = wave_mm16(W1, B, lane);
  frag_store(T, t1, lane);
  WBAR();
  v8f u = wave_mm16(T, W1, lane);
  frag_store(B, u, lane);
  WBAR();

  // v = triu(u)
  for (int idx = (int)lane; idx < UT; idx += 32)
    vbuf[wave][idx] = B[rows_s[idx] * 16 + cols_s[idx]];
  WBAR();

  // dist = v^T Sig_c v   (Sig_c streams from L2: 16 * 74KB resident)
  float acc = 0.f;
  for (int p = (int)lane; p < UT; p += 32) {
    const float* srow = Sg + (size_t)p * UT;
    const float vp = vbuf[wave][p];
    float w = 0.f;
    for (int q = 0; q < UT; ++q) w = fmaf(srow[q], vbuf[wave][q], w);
    acc = fmaf(vp, w, acc);
  }
#pragma unroll
  for (int m = 16; m >= 1; m >>= 1) acc += __shfl_xor(acc, m, 32);

  if (lane == 0) out[(size_t)n * CC + c] = -0.5f * acc + bias[c];
}

// ---------------------------------------------------------------------------
extern "C" void kernel_launch(void* const* d_in, const int* in_sizes, int n_in,
                              void* d_out, int out_size, void* d_ws,
                              size_t ws_size, hipStream_t stream) {
  const float* node_feats = (const float*)d_in[0];  // (N,16,16)
  const float* means      = (const float*)d_in[1];  // (C,16,16)
  const float* sigma      = (const float*)d_in[2];  // (C,136,136)
  const float* bias       = (const float*)d_in[3];  // (C,)
  float* ws = (float*)d_ws;

  float* sqrt_x     = ws;                         // 16*256
  float* inv_sqrt_x = ws + 4096;                  // 16*256
  float* sig        = ws + 8192;                  // 16*136*136
  float* scratch    = sig + (size_t)CC * UT * UT; // 3*16*136*136

  spd_class_prep<<<4, 128, 0, stream>>>(means, sqrt_x, inv_sqrt_x);
  spd_sigma_expm<<<CC, 256, 0, stream>>>(sigma, sig, scratch);

  const int pairs  = NN * CC;          // 320000
  const int blocks = pairs / WPB;      // 80000
  spd_main<<<blocks, 128, 0, stream>>>(node_feats, sqrt_x, inv_sqrt_x, sig,
                                       bias, (float*)d_out);
}